// MultiHeadAttentionOfficial_72267119722914
// MI455X (gfx1250) — compile-verified
//
#include <hip/hip_runtime.h>
#include <hip/hip_bf16.h>

// Problem constants (match reference)
#define H   16
#define DK  64
#define DM  1024     // d_model = H*DK
#define BB  2
#define NN  2048
#define EPS 1e-6f
#define MROWS (BB * NN)   // 4096 token rows

typedef __attribute__((ext_vector_type(2))) float v2f;
typedef __attribute__((ext_vector_type(8))) float v8f;

// D = A(16x4, f32) * B(4x16, f32) + C(16x16, f32)  -> v_wmma_f32_16x16x4_f32
__device__ __forceinline__ v8f wmma4(v2f a, v2f b, v8f c) {
    return __builtin_amdgcn_wmma_f32_16x16x4_f32(
        /*neg_a=*/false, a, /*neg_b=*/false, b,
        /*c_mod=*/(short)0, c, /*reuse_a=*/false, /*reuse_b=*/false);
}

// ---------------------------------------------------------------------------
// One wave computes a (16*MT)-row x 64-col tile of C = A*B (+R).
// A: row-major, lda.  B: BT==false -> B[k*ldb + n]; BT==true -> B[n*ldb + k].
// f32 WMMA fragment layout (ISA 7.12.2):
//   A frag: lane l -> row (l&15), k-pair base 2*(l>>4); VGPR0=k, VGPR1=k+1
//   B frag: lane l -> col (l&15), k-pair base 2*(l>>4); VGPR0=k, VGPR1=k+1
//   C/D:    VGPR j -> row j + 8*(l>>4), col (l&15)
// MT=2 reuses each B fragment across two row tiles (halves B load traffic).
// ---------------------------------------------------------------------------
template <int MT, bool BT, bool RES>
__device__ __forceinline__ void gemm_strip(
    const float* __restrict__ A, int lda,
    const float* __restrict__ Bm, int ldb,
    float* __restrict__ C, long ldc,
    const float* __restrict__ R, long ldr,
    int K)
{
    const int lane = threadIdx.x & 31;
    const int half = lane >> 4;   // 0 or 1
    const int l16  = lane & 15;

    v8f zero = {0.f, 0.f, 0.f, 0.f, 0.f, 0.f, 0.f, 0.f};
    v8f acc[MT][4];
    #pragma unroll
    for (int m = 0; m < MT; ++m)
        #pragma unroll
        for (int t = 0; t < 4; ++t) acc[m][t] = zero;

    const float* arow[MT];
    #pragma unroll
    for (int m = 0; m < MT; ++m)
        arow[m] = A + (long)(m * 16 + l16) * lda + 2 * half;   // A[row, 2*half + k]

    #pragma unroll 2
    for (int k = 0; k < K; k += 4) {
        v2f a[MT];
        #pragma unroll
        for (int m = 0; m < MT; ++m) {
            a[m].x = arow[m][k];
            a[m].y = arow[m][k + 1];
        }
        #pragma unroll
        for (int t = 0; t < 4; ++t) {
            const int n = t * 16 + l16;
            v2f b;
            if (BT) {
                const float* br = Bm + (long)n * ldb + 2 * half;
                b.x = br[k];
                b.y = br[k + 1];
            } else {
                const float* bc = Bm + (long)(k + 2 * half) * ldb + n;
                b.x = bc[0];
                b.y = bc[ldb];
            }
            #pragma unroll
            for (int m = 0; m < MT; ++m)           // B frag reused across MT row tiles
                acc[m][t] = wmma4(a[m], b, acc[m][t]);
        }
    }

    #pragma unroll
    for (int m = 0; m < MT; ++m)
        #pragma unroll
        for (int t = 0; t < 4; ++t)
            #pragma unroll
            for (int j = 0; j < 8; ++j) {
                const int row = m * 16 + j + 8 * half;
                const int col = t * 16 + l16;
                float val = acc[m][t][j];
                if (RES) val += R[(long)row * ldr + col];
                C[(long)row * ldc + col] = val;
            }
}

// --------------------- 1) QKV projections: X @ Wq ---------------------------
// grid (4, 128, 3), block 128.  z selects q/k/v source & destination slab.
__global__ __launch_bounds__(128) void proj_kernel(
    const float* __restrict__ q, const float* __restrict__ k,
    const float* __restrict__ v, const float* __restrict__ Wq,
    float* __restrict__ qh, float* __restrict__ kh, float* __restrict__ vh)
{
    const int wave = threadIdx.x >> 5;
    const long M0 = (long)blockIdx.y * 32;
    const long N0 = ((long)blockIdx.x * 4 + wave) * 64;
    const float* X = (blockIdx.z == 0) ? q : (blockIdx.z == 1) ? k : v;
    float*       P = (blockIdx.z == 0) ? qh : (blockIdx.z == 1) ? kh : vh;
    gemm_strip<2, false, false>(X + M0 * DM, DM, Wq + N0, DM,
                                P + M0 * DM + N0, DM, nullptr, 0, DM);
}

// --------------------- 2) scores = QH @ KH^T per (b,h) ----------------------
// grid (8, 64, 32), block 128.  Writes straight into d_out's attn region.
__global__ __launch_bounds__(128) void scores_kernel(
    const float* __restrict__ qh, const float* __restrict__ kh,
    float* __restrict__ attn)
{
    const int wave = threadIdx.x >> 5;
    const int bh = blockIdx.z;            // b*H + h
    const int b  = bh / H;
    const int h  = bh - b * H;
    const long M0 = (long)blockIdx.y * 32;                 // query rows
    const long N0 = ((long)blockIdx.x * 4 + wave) * 64;    // key cols
    const float* A  = qh + ((long)b * NN + M0) * DM + h * DK;      // lda=DM
    const float* Bm = kh + ((long)b * NN + N0) * DM + h * DK;      // B[n,k], BT
    float* C = attn + ((long)bh * NN + M0) * NN + N0;              // ldc=NN
    gemm_strip<2, true, false>(A, DM, Bm, DM, C, NN, nullptr, 0, DK);
}

// --------------------- 3) softmax over the HEAD axis (source quirk) ---------
__global__ __launch_bounds__(256) void softmax_heads_kernel(float* __restrict__ attn)
{
    const long total = (long)BB * NN * NN;
    long idx = (long)blockIdx.x * blockDim.x + threadIdx.x;   // (b, n, m) flat
    if (idx >= total) return;
    const long b = idx / ((long)NN * NN);
    const long r = idx - b * (long)NN * NN;
    float* p = attn + b * (long)H * NN * NN + r;
    const long hs = (long)NN * NN;

    float vals[H];
    float m = -3.402823466e38f;
    #pragma unroll
    for (int h = 0; h < H; ++h) { vals[h] = p[h * hs]; m = fmaxf(m, vals[h]); }
    float s = 0.f;
    #pragma unroll
    for (int h = 0; h < H; ++h) { vals[h] = __expf(vals[h] - m); s += vals[h]; }
    const float inv = 1.f / s;
    #pragma unroll
    for (int h = 0; h < H; ++h) p[h * hs] = vals[h] * inv;
}

// --------------------- 4) ctx = attn @ VH per (b,h) -------------------------
// grid (1, 16, 32), block 128.  ctx laid out as (B, N, H*DK).
__global__ __launch_bounds__(128) void attnv_kernel(
    const float* __restrict__ attn, const float* __restrict__ vh,
    float* __restrict__ ctx)
{
    const int wave = threadIdx.x >> 5;
    const int bh = blockIdx.z;
    const int b  = bh / H;
    const int h  = bh - b * H;
    const long M0 = ((long)blockIdx.y * 4 + wave) * 32;       // query rows
    const float* A  = attn + ((long)bh * NN + M0) * NN;       // lda=NN
    const float* Bm = vh + (long)b * NN * DM + h * DK;        // B[k,n]=VH[k, h*DK+n]
    float* C = ctx + ((long)b * NN + M0) * DM + h * DK;       // ldc=DM
    gemm_strip<2, false, false>(A, NN, Bm, DM, C, DM, nullptr, 0, NN);
}

// --------------------- 5) out_pre = ctx @ Wfc + residual(q) -----------------
// grid (4, 128), block 128.
__global__ __launch_bounds__(128) void fc_kernel(
    const float* __restrict__ ctx, const float* __restrict__ Wfc,
    const float* __restrict__ resid, float* __restrict__ outp)
{
    const int wave = threadIdx.x >> 5;
    const long M0 = (long)blockIdx.y * 32;
    const long N0 = ((long)blockIdx.x * 4 + wave) * 64;
    gemm_strip<2, false, true>(ctx + M0 * DM, DM, Wfc + N0, DM,
                               outp + M0 * DM + N0, DM,
                               resid + M0 * DM + N0, DM, DM);
}

// --------------------- 6) row LayerNorm -> d_out[0 .. B*N*DM) ---------------
__global__ __launch_bounds__(256) void ln_kernel(
    const float* __restrict__ outp, const float* __restrict__ gamma,
    const float* __restrict__ beta, float* __restrict__ ln)
{
    const int row = blockIdx.x;
    const float* x = outp + (long)row * DM;
    float* y = ln + (long)row * DM;

    float s = 0.f, s2 = 0.f;
    for (int i = threadIdx.x; i < DM; i += blockDim.x) {
        const float t = x[i];
        s += t; s2 += t * t;
    }
    __shared__ float ps[8], ps2[8];
    const int wid  = threadIdx.x >> 5;
    const int lane = threadIdx.x & 31;
    for (int off = 16; off > 0; off >>= 1) {
        s  += __shfl_down(s,  off, 32);
        s2 += __shfl_down(s2, off, 32);
    }
    if (lane == 0) { ps[wid] = s; ps2[wid] = s2; }
    __syncthreads();
    if (wid == 0) {
        s  = (lane < 8) ? ps[lane]  : 0.f;
        s2 = (lane < 8) ? ps2[lane] : 0.f;
        for (int off = 4; off > 0; off >>= 1) {
            s  += __shfl_down(s,  off, 32);
            s2 += __shfl_down(s2, off, 32);
        }
        if (lane == 0) { ps[0] = s; ps2[0] = s2; }
    }
    __syncthreads();
    const float mean = ps[0] * (1.f / DM);
    const float var  = ps2[0] * (1.f / DM) - mean * mean;
    const float rstd = rsqrtf(var + EPS);
    for (int i = threadIdx.x; i < DM; i += blockDim.x)
        y[i] = (x[i] - mean) * rstd * gamma[i] + beta[i];
}

// ---------------------------------------------------------------------------
extern "C" void kernel_launch(void* const* d_in, const int* in_sizes, int n_in,
                              void* d_out, int out_size, void* d_ws, size_t ws_size,
                              hipStream_t stream)
{
    const float* q     = (const float*)d_in[0];
    const float* k     = (const float*)d_in[1];
    const float* v     = (const float*)d_in[2];
    const float* Wq    = (const float*)d_in[3];
    const float* Wfc   = (const float*)d_in[4];
    const float* gamma = (const float*)d_in[5];
    const float* beta  = (const float*)d_in[6];

    float* out  = (float*)d_out;
    float* ln   = out;                                   // (B, N, DM)
    float* attn = out + (long)BB * NN * DM;              // (B, H, N, N)

    const long SLAB = (long)BB * NN * DM;                // 4,194,304 floats
    float* ws   = (float*)d_ws;
    float* qh   = ws;                                    // (B*N, H*DK)
    float* kh   = ws + SLAB;
    float* vh   = ws + 2 * SLAB;
    float* ctx  = ws + 3 * SLAB;                         // (B, N, H*DK)
    float* outp = ws + 4 * SLAB;                         // pre-LN

    const dim3 blk(128);   // 4 wave32 waves

    // 1) QH/KH/VH = {q,k,v} @ Wq
    proj_kernel<<<dim3(4, 128, 3), blk, 0, stream>>>(q, k, v, Wq, qh, kh, vh);
    // 2) scores -> attn region of d_out
    scores_kernel<<<dim3(8, 64, BB * H), blk, 0, stream>>>(qh, kh, attn);
    // 3) softmax across heads, in place
    const long nsm = (long)BB * NN * NN;
    softmax_heads_kernel<<<dim3((unsigned)((nsm + 255) / 256)), dim3(256), 0, stream>>>(attn);
    // 4) ctx = attn @ VH
    attnv_kernel<<<dim3(1, 16, BB * H), blk, 0, stream>>>(attn, vh, ctx);
    // 5) out_pre = ctx @ Wfc + q
    fc_kernel<<<dim3(4, 128), blk, 0, stream>>>(ctx, Wfc, q, outp);
    // 6) LayerNorm -> ln region of d_out
    ln_kernel<<<dim3(MROWS), dim3(256), 0, stream>>>(outp, gamma, beta, ln);
}